// SharedGNN_43241730736593
// MI455X (gfx1250) — compile-verified
//
#include <hip/hip_runtime.h>

typedef __attribute__((ext_vector_type(2))) float v2f;
typedef __attribute__((ext_vector_type(8))) float v8f;

constexpr int N_NODES  = 20000;
constexpr int N_EDGES  = 320000;
constexpr int BATCH    = 8;
constexpr int N_OBS    = 6;
constexpr int N_FIXED  = 10;
constexpr int HID      = 128;
constexpr int ROWS     = BATCH * N_NODES;   // 160000
constexpr int MB_ROWS  = 64;                // GEMM M-tile per block
constexpr int LH_PITCH = HID + 4;           // LDS padding: bank = (4m+k)%64, conflict-free A reads

// ---------------- degree / norm precompute ----------------
__global__ void k_deg_init(float* deg) {
    int i = blockIdx.x * 256 + threadIdx.x;
    if (i < N_NODES) deg[i] = 1.0f;          // self-loop contribution
}

__global__ void k_deg_accum(const int* __restrict__ col, float* deg) {
    int i = blockIdx.x * 256 + threadIdx.x;
    if (i < N_EDGES) atomicAdd(&deg[col[i]], 1.0f);
}

__global__ void k_deg_finalize(float* dinv) {
    int i = blockIdx.x * 256 + threadIdx.x;
    if (i < N_NODES) dinv[i] = rsqrtf(dinv[i]);   // deg >= 1 always (self-loops)
}

__global__ void k_norm(const int* __restrict__ row, const int* __restrict__ col,
                       const float* __restrict__ dinv, float* __restrict__ normv) {
    int i = blockIdx.x * 256 + threadIdx.x;
    if (i < N_EDGES) normv[i] = dinv[row[i]] * dinv[col[i]];
}

// ---------------- pack Wg into paired-K layout for single-b64 B fragments ----------------
// Wp[l][k/2][n][{0,1}] = Wg[l][k][n], Wg[l][k+1][n]
__global__ void k_pack_w(const float* __restrict__ Wg, float* __restrict__ Wp) {
    int gid = blockIdx.x * 256 + threadIdx.x;   // 3*128*128 = 49152
    if (gid >= 3 * HID * HID) return;
    int l   = gid / (HID * HID);
    int rem = gid - l * HID * HID;
    int kk  = rem >> 7;
    int n   = rem & (HID - 1);
    Wp[l * HID * HID + (kk >> 1) * (2 * HID) + n * 2 + (kk & 1)] = Wg[gid];
}

// ---------------- input projection: h = [x | fixed] @ W_in + b_in ----------------
__global__ void k_input_proj(const float* __restrict__ x, const float* __restrict__ fixedf,
                             const float* __restrict__ W_in, const float* __restrict__ b_in,
                             float* __restrict__ h) {
    int gid = blockIdx.x * 256 + threadIdx.x;   // ROWS*HID threads exactly
    int c = gid & (HID - 1);
    int r = gid >> 7;
    int node = r % N_NODES;
    const float* xr = x + r * N_OBS;
    const float* fr = fixedf + node * N_FIXED;
    float acc = b_in[c];
#pragma unroll
    for (int k = 0; k < N_OBS; ++k)   acc += xr[k] * W_in[k * HID + c];
#pragma unroll
    for (int k = 0; k < N_FIXED; ++k) acc += fr[k] * W_in[(N_OBS + k) * HID + c];
    h[gid] = acc;
}

// ---------------- t = h @ W  via V_WMMA_F32_16X16X4_F32 ----------------
// block = 256 threads = 8 wave32s; 64-row M-tile async-copied into LDS;
// wave w owns output cols [16w,16w+16); 4 accumulators reuse each B fragment.
__global__ __launch_bounds__(256) void k_gemm_wmma(const float* __restrict__ h,
                                                   const float* __restrict__ Wp,
                                                   float* __restrict__ t) {
    __shared__ float lh[MB_ROWS * LH_PITCH];
    const int tid  = threadIdx.x;
    const int wave = tid >> 5;
    const int lane = tid & 31;
    const int rowbase = blockIdx.x * MB_ROWS;   // ROWS % 64 == 0, no tail

    // Async global->LDS stage of the 64x128 tile (contiguous 32KB in global).
    // 2048 x B128 chunks; LDS dest honors the +4-float row padding.
    {
        const float* gsrc = h + rowbase * HID;
#pragma unroll
        for (int j = 0; j < 8; ++j) {
            int i  = tid + j * 256;          // 16B chunk id
            int r  = i >> 5;                 // tile row
            int c4 = i & 31;                 // float4 within row
            unsigned lds_addr = (unsigned)(unsigned long long)&lh[r * LH_PITCH + c4 * 4];
            const float* g = gsrc + i * 4;
            asm volatile("global_load_async_to_lds_b128 %0, %1, off"
                         :: "v"(lds_addr), "v"(g) : "memory");
        }
        asm volatile("s_wait_asynccnt 0x0" ::: "memory");
    }
    __syncthreads();

    const int m16  = lane & 15;
    const int half = lane >> 4;          // 0: lanes 0-15, 1: lanes 16-31
    const int koff = half * 2;           // A/B: upper lanes hold K+2,K+3
    const int ncol = wave * 16 + m16;

    v8f acc[4] = {};
    for (int k = 0; k < HID; k += 4) {
        const int kk = k + koff;                               // even
        v2f b = *(const v2f*)(Wp + ((kk >> 1) << 8) + (ncol << 1));  // one b64 load
#pragma unroll
        for (int mt = 0; mt < 4; ++mt) {                       // reuse B across M sub-tiles
            v2f a = *(const v2f*)&lh[(mt * 16 + m16) * LH_PITCH + kk];
            acc[mt] = __builtin_amdgcn_wmma_f32_16x16x4_f32(
                false, a, false, b, (short)0, acc[mt], false, false);
        }
    }

    // D layout: VGPR j -> row j (lanes 0-15) / row j+8 (lanes 16-31), col = lane&15
#pragma unroll
    for (int mt = 0; mt < 4; ++mt) {
#pragma unroll
        for (int j = 0; j < 8; ++j) {
            int r = rowbase + mt * 16 + j + half * 8;
            t[r * HID + ncol] = acc[mt][j];
        }
    }
}

// ---------------- agg = t * dinv(node)^2  (self-loop term) ----------------
__global__ void k_selfinit(const float* __restrict__ t, const float* __restrict__ dinv,
                           float* __restrict__ agg) {
    int gid = blockIdx.x * 256 + threadIdx.x;
    int r = gid >> 7;
    int node = r % N_NODES;
    float d = dinv[node];
    agg[gid] = t[gid] * d * d;
}

// ---------------- edge scatter: agg[col] += t[row] * norm ----------------
// one wave32 per (edge, batch); 4 channels per lane; f32 global atomics (L2-resident)
__global__ void k_scatter(const int* __restrict__ erow, const int* __restrict__ ecol,
                          const float* __restrict__ normv, const float* __restrict__ t,
                          float* __restrict__ agg) {
    int task = blockIdx.x * 8 + (threadIdx.x >> 5);   // 2.56M wave-tasks
    int lane = threadIdx.x & 31;
    int e = task >> 3;
    int b = task & 7;
    if (e >= N_EDGES) return;                         // uniform per wave
    int rs = erow[e], cd = ecol[e];
    float w = normv[e];
    const float* src = t   + (b * N_NODES + rs) * HID;
    float*       dst = agg + (b * N_NODES + cd) * HID;
#pragma unroll
    for (int j = 0; j < 4; ++j) {
        int c = lane + j * 32;
        atomicAdd(&dst[c], src[c] * w);
    }
}

// ---------------- h = relu(agg + bg[l]) ----------------
__global__ void k_biasrelu(const float* __restrict__ agg, const float* __restrict__ bgl,
                           float* __restrict__ out) {
    int gid = blockIdx.x * 256 + threadIdx.x;
    int c = gid & (HID - 1);
    float v = agg[gid] + bgl[c];
    out[gid] = v > 0.0f ? v : 0.0f;
}

extern "C" void kernel_launch(void* const* d_in, const int* in_sizes, int n_in,
                              void* d_out, int out_size, void* d_ws, size_t ws_size,
                              hipStream_t stream) {
    const float* x      = (const float*)d_in[0];
    const float* fixedf = (const float*)d_in[1];
    const float* W_in   = (const float*)d_in[2];
    const float* b_in   = (const float*)d_in[3];
    const float* Wg     = (const float*)d_in[4];
    const float* bg     = (const float*)d_in[5];
    const int*   ei     = (const int*)d_in[6];
    const int* erow = ei;
    const int* ecol = ei + N_EDGES;
    float* out = (float*)d_out;

    float* ws   = (float*)d_ws;
    float* dinv = ws;                                   // 20000
    float* nrm  = ws + N_NODES;                         // 320000
    float* Wp   = ws + N_NODES + N_EDGES;               // 3*128*128 packed weights
    float* bufA = Wp + 3 * HID * HID;                   // h / agg (ROWS*HID)
    float* bufB = bufA + ROWS * HID;                    // t       (ROWS*HID)

    const int elemBlocks = (ROWS * HID) / 256;          // 80000
    const int edgeBlocks = (N_EDGES + 255) / 256;

    k_deg_init    <<<(N_NODES + 255) / 256, 256, 0, stream>>>(dinv);
    k_deg_accum   <<<edgeBlocks, 256, 0, stream>>>(ecol, dinv);
    k_deg_finalize<<<(N_NODES + 255) / 256, 256, 0, stream>>>(dinv);
    k_norm        <<<edgeBlocks, 256, 0, stream>>>(erow, ecol, dinv, nrm);
    k_pack_w      <<<(3 * HID * HID + 255) / 256, 256, 0, stream>>>(Wg, Wp);

    k_input_proj<<<elemBlocks, 256, 0, stream>>>(x, fixedf, W_in, b_in, bufA);

    for (int l = 0; l < 3; ++l) {
        k_gemm_wmma<<<ROWS / MB_ROWS, 256, 0, stream>>>(bufA, Wp + l * HID * HID, bufB);
        k_selfinit <<<elemBlocks, 256, 0, stream>>>(bufB, dinv, bufA);
        k_scatter  <<<N_EDGES, 256, 0, stream>>>(erow, ecol, nrm, bufB, bufA);
        k_biasrelu <<<elemBlocks, 256, 0, stream>>>(bufA, bg + l * HID,
                                                    (l == 2) ? out : bufA);
    }
}